// CombinedHiddenPRADAEncoder_369367188151
// MI455X (gfx1250) — compile-verified
//
#include <hip/hip_runtime.h>

typedef __attribute__((ext_vector_type(2))) float v2f;
typedef __attribute__((ext_vector_type(8))) float v8f;

#define THREADS 256

// ---------------- degree / dinv ----------------
__global__ void k_init_deg(float* deg, int n) {
  int i = blockIdx.x * blockDim.x + threadIdx.x;
  if (i < n) deg[i] = 1.0f;  // self loop
}

__global__ void k_count_deg(const int* __restrict__ dst, float* deg, int e) {
  int i = blockIdx.x * blockDim.x + threadIdx.x;
  if (i < e) atomicAdd(&deg[dst[i]], 1.0f);
}

__global__ void k_rsqrt(float* d, int n) {
  int i = blockIdx.x * blockDim.x + threadIdx.x;
  if (i < n) d[i] = rsqrtf(d[i]);
}

// ---------------- GEMM1: hs = (X[ M x128 ] @ W[128x64]) * dinv[m]; also acc init ----
// block = 128 threads (4 waves). Wave w computes 16x16 tile at (blockIdx*16, w*16).
__global__ void k_gemm1(const float* __restrict__ X, const float* __restrict__ W,
                        const float* __restrict__ dinv, float* __restrict__ hs,
                        float* __restrict__ acc, int Nn) {
  __shared__ float sW[128 * 64];   // 32 KB
  __shared__ float sX[16 * 128];   // 8 KB
  const int tid = threadIdx.x;
  const int m0 = blockIdx.x * 16;

  for (int i = tid * 4; i < 128 * 64; i += 128 * 4)
    *(float4*)(sW + i) = *(const float4*)(W + i);
  for (int i = tid * 4; i < 16 * 128; i += 128 * 4) {
    int r = i >> 7, cc = i & 127;
    int m = m0 + r; if (m >= Nn) m = Nn - 1;
    *(float4*)(sX + i) = *(const float4*)(X + (size_t)m * 128 + cc);
  }
  __syncthreads();

  const int wave = tid >> 5;
  const int lane = tid & 31;
  const int row  = lane & 15;   // M (for A/C) or N (for B)
  const int kh   = lane >> 4;   // half selector
  const int n0   = wave * 16;

  v8f c = {};
  for (int k0 = 0; k0 < 128; k0 += 4) {
    const int ka = k0 + kh * 2;
    v2f a, b;
    // A 16x4 f32: lanes 0-15 -> K=k0,k0+1 ; lanes 16-31 -> K=k0+2,k0+3 ; M=lane%16
    a.x = sX[row * 128 + ka];
    a.y = sX[row * 128 + ka + 1];
    // B 4x16 f32: VGPR0 lanes 0-15 -> K=k0 row, lanes 16-31 -> K=k0+2 ; N=lane%16
    b.x = sW[ka * 64 + n0 + row];
    b.y = sW[(ka + 1) * 64 + n0 + row];
    c = __builtin_amdgcn_wmma_f32_16x16x4_f32(false, a, false, b, (short)0, c, false, false);
  }
  // C/D 16x16: VGPR j, lanes 0-15 -> M=j ; lanes 16-31 -> M=j+8 ; N=lane%16
  for (int j = 0; j < 8; ++j) {
    int m = m0 + j + kh * 8;
    if (m < Nn) {
      float v = c[j] * dinv[m];
      size_t o = (size_t)m * 64 + n0 + row;
      hs[o] = v;
      acc[o] = v;  // self-loop contribution
    }
  }
}

// ---------------- GEMM2: ms = (H @ Wm)*dinv, vs = (H @ Wv)*dinv ----------------
// waves 0,1 -> Wm (out base msv), waves 2,3 -> Wv (out base msv + Nn*32)
__global__ void k_gemm2(const float* __restrict__ H, const float* __restrict__ Wm,
                        const float* __restrict__ Wv, const float* __restrict__ dinv,
                        float* __restrict__ msv, int Nn) {
  __shared__ float sW[2 * 64 * 32];  // 16 KB (Wm then Wv)
  __shared__ float sH[16 * 64];      // 4 KB
  const int tid = threadIdx.x;
  const int m0 = blockIdx.x * 16;

  for (int i = tid * 4; i < 64 * 32; i += 128 * 4) {
    *(float4*)(sW + i)        = *(const float4*)(Wm + i);
    *(float4*)(sW + 2048 + i) = *(const float4*)(Wv + i);
  }
  for (int i = tid * 4; i < 16 * 64; i += 128 * 4) {
    int r = i >> 6, cc = i & 63;
    int m = m0 + r; if (m >= Nn) m = Nn - 1;
    *(float4*)(sH + i) = *(const float4*)(H + (size_t)m * 64 + cc);
  }
  __syncthreads();

  const int wave = tid >> 5;
  const int lane = tid & 31;
  const int row  = lane & 15;
  const int kh   = lane >> 4;
  const int n0   = (wave & 1) * 16;
  const float* sWs = sW + (wave >> 1) * 2048;

  v8f c = {};
  for (int k0 = 0; k0 < 64; k0 += 4) {
    const int ka = k0 + kh * 2;
    v2f a, b;
    a.x = sH[row * 64 + ka];
    a.y = sH[row * 64 + ka + 1];
    b.x = sWs[ka * 32 + n0 + row];
    b.y = sWs[(ka + 1) * 32 + n0 + row];
    c = __builtin_amdgcn_wmma_f32_16x16x4_f32(false, a, false, b, (short)0, c, false, false);
  }
  float* outb = msv + (size_t)(wave >> 1) * (size_t)Nn * 32;
  for (int j = 0; j < 8; ++j) {
    int m = m0 + j + kh * 8;
    if (m < Nn) outb[(size_t)m * 32 + n0 + row] = c[j] * dinv[m];
  }
}

// ---------------- edge scatter, layer 1 (64 ch): 16 threads/edge, float4 each ----
__global__ void k_scatter1(const int* __restrict__ src, const int* __restrict__ dst,
                           const float* __restrict__ hs, float* __restrict__ acc,
                           long long total) {
  long long idx = (long long)blockIdx.x * blockDim.x + threadIdx.x;
  if (idx >= total) return;
  int e = (int)(idx >> 4);
  int q = (int)(idx & 15);
  int s = src[e], d = dst[e];
  float4 v = *(const float4*)(hs + (size_t)s * 64 + q * 4);
  float* p = acc + (size_t)d * 64 + q * 4;
  atomicAdd(p + 0, v.x); atomicAdd(p + 1, v.y);
  atomicAdd(p + 2, v.z); atomicAdd(p + 3, v.w);
}

// ---------------- edge scatter, layer 2 (2 heads x 32 ch): 16 threads/edge ----
__global__ void k_scatter2(const int* __restrict__ src, const int* __restrict__ dst,
                           const float* __restrict__ msv, float* __restrict__ accmv,
                           int Nn, long long total) {
  long long idx = (long long)blockIdx.x * blockDim.x + threadIdx.x;
  if (idx >= total) return;
  int e = (int)(idx >> 4);
  int q = (int)(idx & 15);
  int s = src[e], d = dst[e];
  int mat = q >> 3;          // 0 = mean head, 1 = logvar head
  int q4  = (q & 7) * 4;
  const float* base = msv   + (size_t)mat * (size_t)Nn * 32;
  float*      abase = accmv + (size_t)mat * (size_t)Nn * 32;
  float4 v = *(const float4*)(base + (size_t)s * 32 + q4);
  float* p = abase + (size_t)d * 32 + q4;
  atomicAdd(p + 0, v.x); atomicAdd(p + 1, v.y);
  atomicAdd(p + 2, v.z); atomicAdd(p + 3, v.w);
}

// ---------------- h = tanh(dinv * acc + b1), in place ----------------
__global__ void k_tanh(float* acc, const float* __restrict__ dinv,
                       const float* __restrict__ b, int total) {
  int i = blockIdx.x * blockDim.x + threadIdx.x;
  if (i < total) {
    int n = i >> 6, cc = i & 63;
    acc[i] = tanhf(dinv[n] * acc[i] + b[cc]);
  }
}

__global__ void k_copy4(const float4* __restrict__ a, float4* __restrict__ b, int cnt) {
  int i = blockIdx.x * blockDim.x + threadIdx.x;
  if (i < cnt) b[i] = a[i];
}

// ---------------- reparameterize + write (z, mean, logvar) ----------------
__global__ void k_final(const float* __restrict__ accmv, const float* __restrict__ dinv,
                        const float* __restrict__ bm, const float* __restrict__ bv,
                        const float* __restrict__ noise, float* __restrict__ out, int Nn) {
  int i = blockIdx.x * blockDim.x + threadIdx.x;
  int total = Nn * 32;
  if (i >= total) return;
  int n = i >> 5, cc = i & 31;
  float dm = dinv[n];
  float mean = dm * accmv[i] + bm[cc];
  float logv = dm * accmv[(size_t)total + i] + bv[cc];
  float z = noise[i] * expf(0.5f * logv) + mean;
  out[i] = z;
  out[total + i] = mean;
  out[2 * total + i] = logv;
}

extern "C" void kernel_launch(void* const* d_in, const int* in_sizes, int n_in,
                              void* d_out, int out_size, void* d_ws, size_t ws_size,
                              hipStream_t stream) {
  const float* X     = (const float*)d_in[0];
  const int*   edge  = (const int*)d_in[1];
  const float* W1    = (const float*)d_in[2];
  const float* b1    = (const float*)d_in[3];
  const float* Wm    = (const float*)d_in[4];
  const float* bm    = (const float*)d_in[5];
  const float* Wv    = (const float*)d_in[6];
  const float* bv    = (const float*)d_in[7];
  const float* noise = (const float*)d_in[8];

  const int Nn = in_sizes[0] / 128;
  const int E  = in_sizes[1] / 2;
  const int* src = edge;
  const int* dst = edge + E;

  float* ws = (float*)d_ws;
  size_t off = 0;
  auto alloc = [&](size_t cnt) {
    float* p = ws + off;
    off += (cnt + 15) & ~(size_t)15;
    return p;
  };
  float* dinv = alloc((size_t)Nn);
  float* R1   = alloc((size_t)Nn * 64);  // hs1, then ms||vs
  float* R2   = alloc((size_t)Nn * 64);  // acc1 -> h, then accm||accv
  (void)ws_size; (void)n_in; (void)out_size;

  const long long e16 = (long long)E * 16;
  const int gN   = (Nn + THREADS - 1) / THREADS;
  const int gE   = (E + THREADS - 1) / THREADS;
  const int gE16 = (int)((e16 + THREADS - 1) / THREADS);
  const int gMT  = (Nn + 15) / 16;
  const int gN64 = (Nn * 64 + THREADS - 1) / THREADS;
  const int gN32 = (Nn * 32 + THREADS - 1) / THREADS;
  const int gCp  = (Nn * 16 + THREADS - 1) / THREADS;

  // degree -> dinv (dst-degree incl. self loop, matching reference)
  k_init_deg <<<gN,   THREADS, 0, stream>>>(dinv, Nn);
  k_count_deg<<<gE,   THREADS, 0, stream>>>(dst, dinv, E);
  k_rsqrt    <<<gN,   THREADS, 0, stream>>>(dinv, Nn);
  // layer 1
  k_gemm1    <<<gMT,  128,     0, stream>>>(X, W1, dinv, R1, R2, Nn);
  k_scatter1 <<<gE16, THREADS, 0, stream>>>(src, dst, R1, R2, e16);
  k_tanh     <<<gN64, THREADS, 0, stream>>>(R2, dinv, b1, Nn * 64);
  // layer 2 (mean + logvar heads)
  k_gemm2    <<<gMT,  128,     0, stream>>>(R2, Wm, Wv, dinv, R1, Nn);
  k_copy4    <<<gCp,  THREADS, 0, stream>>>((const float4*)R1, (float4*)R2, Nn * 16);
  k_scatter2 <<<gE16, THREADS, 0, stream>>>(src, dst, R1, R2, Nn, e16);
  // reparameterize + outputs
  k_final    <<<gN32, THREADS, 0, stream>>>(R2, dinv, bm, bv, noise, (float*)d_out, Nn);
}